// ObsEncode_32160715112724
// MI455X (gfx1250) — compile-verified
//
#include <hip/hip_runtime.h>
#include <math.h>

typedef __attribute__((ext_vector_type(2))) float v2f;
typedef __attribute__((ext_vector_type(8))) float v8f;

#define NA    16      // agents per batch element
#define OTH   15      // others
#define HEADS 6
#define DKD   16
#define DOBS  352     // 4 + 22*15 + 3*6
#define H1    32      // hidden width
#define SCALE_F 0.125f

__device__ __forceinline__ v8f wmma4(v2f a, v2f b, v8f c) {
  // D = A(16x4 f32) * B(4x16 f32) + C(16x16 f32)
  return __builtin_amdgcn_wmma_f32_16x16x4_f32(false, a, false, b, (short)0, c,
                                               false, false);
}

// First MLP layer: X(16x22, as 6 A-frags padded to K=24) * W1^T(->16x32) + b1, relu.
// Result written to per-wave LDS scratch in C layout (row-major 16x32).
__device__ __forceinline__ void mlp_l1(const v2f af[6], const float* w1p,
                                       const float* b1, float* myhid, int lane) {
  const int hi = lane >> 4, lr = lane & 15;
#pragma unroll
  for (int nt = 0; nt < 2; ++nt) {
    const float bias = b1[nt * 16 + lr];
    v8f acc = {bias, bias, bias, bias, bias, bias, bias, bias};
#pragma unroll
    for (int s = 0; s < 6; ++s) {
      v2f bf = *(const v2f*)&w1p[(nt * 16 + lr) * 24 + 4 * s + 2 * hi];
      acc = wmma4(af[s], bf, acc);
    }
#pragma unroll
    for (int r = 0; r < 8; ++r)
      myhid[(r + 8 * hi) * H1 + nt * 16 + lr] = fmaxf(acc[r], 0.0f);
  }
}

__launch_bounds__(512, 1)
__global__ void obs_attn_kernel(const float* __restrict__ obs,
                                const float* __restrict__ skill,
                                const float* __restrict__ ks_w1, const float* __restrict__ ks_b1,
                                const float* __restrict__ ks_w2, const float* __restrict__ ks_b2,
                                const float* __restrict__ qo_w1, const float* __restrict__ qo_b1,
                                const float* __restrict__ qo_w2, const float* __restrict__ qo_b2,
                                const float* __restrict__ vo_w1, const float* __restrict__ vo_b1,
                                const float* __restrict__ vo_w2, const float* __restrict__ vo_b2,
                                float* __restrict__ out1, float* __restrict__ rel) {
  __shared__ float lds[38848];          // 155,392 bytes
  float* w1q = lds;                     // [32][24] K-padded
  float* w1v = w1q + 768;               // [32][24]
  float* w1k = w1v + 768;               // [32][24]
  float* w2q = w1k + 768;               // [96][32]
  float* w2k = w2q + 3072;              // [96][32]
  float* w2v = w2k + 3072;              // [64][32]
  float* b1q = w2v + 2048;              // 32
  float* b1v = b1q + 32;                // 32
  float* b1k = b1v + 32;                // 32
  float* b2q = b1k + 32;                // 96
  float* b2k = b2q + 96;                // 96
  float* b2v = b2k + 96;                // 64
  float* skl = b2v + 64;                // 96  (16 agents x 6 heads)
  float* hid = skl + 96;                // 16 waves x 512 (16x32 scratch each)
  float* kc  = hid + 8192;              // 16 x 16    combined k
  float* qc  = kc + 256;                // 15 x 16x16 combined q
  float* vvs = qc + 3840;               // 15 x 16x64 v
  float* sc  = vvs + 15360;             // 16 x 16    logits/probs

  const int tid = threadIdx.x;
  const int b = blockIdx.x;

  // ---- stage weights (zero-padded K=24 for layer 1) + skill tile ----
  for (int i = tid; i < 768; i += 512) {
    const int r = i / 24, c = i % 24;
    const bool ok = (c < 22);
    w1q[i] = ok ? qo_w1[r * 22 + c] : 0.f;
    w1v[i] = ok ? vo_w1[r * 22 + c] : 0.f;
    w1k[i] = ok ? ks_w1[r * 22 + c] : 0.f;
  }
  for (int i = tid; i < 3072; i += 512) { w2q[i] = qo_w2[i]; w2k[i] = ks_w2[i]; }
  for (int i = tid; i < 2048; i += 512) w2v[i] = vo_w2[i];
  if (tid < 32) { b1q[tid] = qo_b1[tid]; b1v[tid] = vo_b1[tid]; b1k[tid] = ks_b1[tid]; }
  if (tid < 96) { b2q[tid] = qo_b2[tid]; b2k[tid] = ks_b2[tid];
                  skl[tid] = skill[(size_t)b * 96 + tid]; }
  if (tid < 64) b2v[tid] = vo_b2[tid];
  __syncthreads();

  // ---- phase 1: WMMA MLPs. wave o<15 -> others[o] (qo+vo); wave 15 -> self (ks) ----
  const int wave = tid >> 5;
  const int lane = tid & 31;
  const int hi = lane >> 4;
  const int lr = lane & 15;
  float* myhid = hid + wave * 512;
  const float* obsb = obs + (size_t)b * NA * DOBS;

  // Load X tile into A-fragment layout: lane holds row M=lr, K pair (4s+2hi, +1).
  v2f af[6];
  const v2f z2 = {0.f, 0.f};
  if (wave < OTH) {
    const float* xrow = obsb + lr * DOBS + 4 + 22 * wave;   // even offset -> 8B aligned
#pragma unroll
    for (int s = 0; s < 6; ++s) {
      const int c = 4 * s + 2 * hi;
      af[s] = (c < 22) ? *(const v2f*)(xrow + c) : z2;
    }
  } else {
    const float* xrow = obsb + lr * DOBS;                   // self: [0:4] ++ [334:352]
#pragma unroll
    for (int s = 0; s < 6; ++s) {
      const int c = 4 * s + 2 * hi;
      if (c < 4)       af[s] = *(const v2f*)(xrow + c);
      else if (c < 22) af[s] = *(const v2f*)(xrow + 330 + c);
      else             af[s] = z2;
    }
  }

  const float* w1  = (wave < OTH) ? w1q : w1k;
  const float* w2  = (wave < OTH) ? w2q : w2k;
  const float* bb1 = (wave < OTH) ? b1q : b1k;
  const float* bb2 = (wave < OTH) ? b2q : b2k;

  // layer1 (q/k path), re-layout hidden C->A via LDS scratch
  mlp_l1(af, w1, bb1, myhid, lane);
  v2f hf[8];
#pragma unroll
  for (int s = 0; s < 8; ++s)
    hf[s] = *(const v2f*)&myhid[lr * H1 + 4 * s + 2 * hi];

  // layer2: 6 head tiles, fused skill-combine (C layout: row = r+8*hi, col = lr)
  v8f qsum = {0, 0, 0, 0, 0, 0, 0, 0};
#pragma unroll
  for (int h = 0; h < HEADS; ++h) {
    const float bias = bb2[h * 16 + lr];
    v8f acc = {bias, bias, bias, bias, bias, bias, bias, bias};
#pragma unroll
    for (int s = 0; s < 8; ++s) {
      v2f bf = *(const v2f*)&w2[(h * 16 + lr) * H1 + 4 * s + 2 * hi];
      acc = wmma4(hf[s], bf, acc);
    }
#pragma unroll
    for (int r = 0; r < 8; ++r)
      qsum[r] += skl[(r + 8 * hi) * HEADS + h] * acc[r];
  }
  if (wave < OTH) {
#pragma unroll
    for (int r = 0; r < 8; ++r) qc[wave * 256 + (r + 8 * hi) * 16 + lr] = qsum[r];
  } else {
#pragma unroll
    for (int r = 0; r < 8; ++r) kc[(r + 8 * hi) * 16 + lr] = qsum[r];
  }

  // v path (others only): separate layer1 weights, 64-wide layer2, no combine
  if (wave < OTH) {
    mlp_l1(af, w1v, b1v, myhid, lane);
#pragma unroll
    for (int s = 0; s < 8; ++s)
      hf[s] = *(const v2f*)&myhid[lr * H1 + 4 * s + 2 * hi];
#pragma unroll
    for (int nt = 0; nt < 4; ++nt) {
      const float bias = b2v[nt * 16 + lr];
      v8f acc = {bias, bias, bias, bias, bias, bias, bias, bias};
#pragma unroll
      for (int s = 0; s < 8; ++s) {
        v2f bf = *(const v2f*)&w2v[(nt * 16 + lr) * H1 + 4 * s + 2 * hi];
        acc = wmma4(hf[s], bf, acc);
      }
#pragma unroll
      for (int r = 0; r < 8; ++r)
        vvs[wave * 1024 + (r + 8 * hi) * 64 + nt * 16 + lr] = acc[r];
    }
  }
  __syncthreads();

  // ---- phase 2: attention epilogue ----
  if (tid < NA * OTH) {
    const int n = tid / OTH, o = tid % OTH;
    float d = 0.f;
#pragma unroll
    for (int dd = 0; dd < DKD; ++dd)
      d += kc[n * 16 + dd] * qc[o * 256 + n * 16 + dd];
    sc[n * 16 + o] = d * SCALE_F;
  }
  __syncthreads();
  if (tid < NA) {
    const int n = tid;
    float m = -1e30f;
    for (int o = 0; o < OTH; ++o) m = fmaxf(m, sc[n * 16 + o]);
    float e[OTH], s = 0.f;
    for (int o = 0; o < OTH; ++o) { e[o] = expf(sc[n * 16 + o] - m); s += e[o]; }
    const float inv = 1.f / s;
    for (int o = 0; o < OTH; ++o) {
      const float p = e[o] * inv;
      sc[n * 16 + o] = p;
      rel[(size_t)b * (NA * OTH) + n * OTH + o] = p;   // relation (bs, n_a, 15)
    }
  }
  __syncthreads();
  for (int idx = tid; idx < 64 * NA; idx += 512) {      // out (bs, 64, n_a)
    const int d = idx >> 4, a = idx & 15;
    float s = 0.f;
#pragma unroll
    for (int o = 0; o < OTH; ++o)
      s += sc[a * 16 + o] * vvs[o * 1024 + a * 64 + d];
    out1[(size_t)b * 1024 + d * 16 + a] = s;
  }
}

extern "C" void kernel_launch(void* const* d_in, const int* in_sizes, int n_in,
                              void* d_out, int out_size, void* d_ws, size_t ws_size,
                              hipStream_t stream) {
  (void)n_in; (void)out_size; (void)d_ws; (void)ws_size;
  const float* obs   = (const float*)d_in[0];
  const float* skill = (const float*)d_in[1];
  const float* ks_w1 = (const float*)d_in[2];
  const float* ks_b1 = (const float*)d_in[3];
  const float* ks_w2 = (const float*)d_in[4];
  const float* ks_b2 = (const float*)d_in[5];
  const float* qo_w1 = (const float*)d_in[6];
  const float* qo_b1 = (const float*)d_in[7];
  const float* qo_w2 = (const float*)d_in[8];
  const float* qo_b2 = (const float*)d_in[9];
  const float* vo_w1 = (const float*)d_in[10];
  const float* vo_b1 = (const float*)d_in[11];
  const float* vo_w2 = (const float*)d_in[12];
  const float* vo_b2 = (const float*)d_in[13];

  const int bs = in_sizes[0] / (NA * DOBS);   // 8192
  float* out1 = (float*)d_out;                       // (bs, 64, 16)
  float* rel  = out1 + (size_t)bs * 64 * NA;         // (bs, 16, 15)

  obs_attn_kernel<<<bs, 512, 0, stream>>>(obs, skill, ks_w1, ks_b1, ks_w2, ks_b2,
                                          qo_w1, qo_b1, qo_w2, qo_b2,
                                          vo_w1, vo_b1, vo_w2, vo_b2, out1, rel);
}